// DepthWiseConv_AConnect_73667279061372
// MI455X (gfx1250) — compile-verified
//
#include <hip/hip_runtime.h>
#include <cstdint>
#include <cstddef>

// ---------------------------------------------------------------------------
// Depthwise 3x3 conv (VALID) with per-pool noisy weights + 8-bit LQuant.
//   X    : [64,56,56,256] f32 (NHWC)
//   W    : [3,3,256,1]    f32
//   bias : [256]          f32
//   Werr : [8,3,3,256,1]  f32
//   Berr : [8,256]        f32
//   out  : [64,54,54,256] f32,  pool p = n>>3
// Memory-bound (AI ~2 FLOP/B): stream X rows HBM->LDS once via CDNA5 async
// DMA (ASYNCcnt), ring buffer of 4 row slots, compute with VALU FMAs.
// ---------------------------------------------------------------------------

#define HDIM   56
#define WDIM   56
#define CDIM   256
#define HOUT   54
#define WOUT   54
#define CGRP   64                  // channels per block
#define NCGRP  (CDIM / CGRP)       // 4
#define ROWF   (WDIM * CGRP)       // 3584 floats per LDS row slot
#define NSLOT  4
#define STRIPS 2
#define RSTRIP 27                  // output rows per strip (2*27 = 54)

__device__ __forceinline__ void async_b128(uint32_t lds, const float* g) {
    unsigned long long ga = (unsigned long long)(uintptr_t)g;
    asm volatile("global_load_async_to_lds_b128 %0, %1, off"
                 :: "v"(lds), "v"(ga) : "memory");
}
__device__ __forceinline__ void async_b64(uint32_t lds, const float* g) {
    unsigned long long ga = (unsigned long long)(uintptr_t)g;
    asm volatile("global_load_async_to_lds_b64 %0, %1, off"
                 :: "v"(lds), "v"(ga) : "memory");
}
__device__ __forceinline__ void wait_async_le4() {
    asm volatile("s_wait_asynccnt 0x4" ::: "memory");
}
__device__ __forceinline__ void wait_async_0() {
    asm volatile("s_wait_asynccnt 0x0" ::: "memory");
}

__device__ __forceinline__ float quant8(float x) {
    // lquant(x, 8): round(clip(x,-1,1)*127)/127, round-to-nearest-even
    float c = fminf(fmaxf(x, -1.0f), 1.0f);
    return rintf(c * 127.0f) / 127.0f;
}

__global__ __launch_bounds__(256)
void dwconv_aconnect_kernel(const float* __restrict__ X,
                            const float* __restrict__ W,
                            const float* __restrict__ bias,
                            const float* __restrict__ Werr,
                            const float* __restrict__ Berr,
                            float* __restrict__ out)
{
    __shared__ float smem[NSLOT * ROWF];   // 57,344 B ring buffer (4 input rows)

    const int tid   = threadIdx.x;         // 0..255, 8 waves
    const int strip = blockIdx.x;          // 0..1
    const int cg    = blockIdx.y;          // 0..3  channel group of 64
    const int n     = blockIdx.z;          // 0..63 image
    const int p     = n >> 3;              // pool index

    const uint32_t lds_base = (uint32_t)(uintptr_t)(&smem[0]);

    // -------- per-thread noisy weights (float4 over 4 channels) ------------
    const int c4 = tid & 15;               // float4 index within 64-ch group
    const int cq = cg * 16 + c4;           // float4 index within full C=256

    const float4* Wq  = reinterpret_cast<const float4*>(W);
    const float4* WEq = reinterpret_cast<const float4*>(Werr + (size_t)p * 9 * CDIM);
    float4 wv[9];
#pragma unroll
    for (int t = 0; t < 9; ++t) {
        float4 w0 = Wq [t * (CDIM / 4) + cq];
        float4 e0 = WEq[t * (CDIM / 4) + cq];
        wv[t].x = w0.x * e0.x; wv[t].y = w0.y * e0.y;
        wv[t].z = w0.z * e0.z; wv[t].w = w0.w * e0.w;
    }
    float4 b0 = reinterpret_cast<const float4*>(bias)[cq];
    float4 be = reinterpret_cast<const float4*>(Berr + (size_t)p * CDIM)[cq];
    float4 bv;
    bv.x = b0.x * be.x; bv.y = b0.y * be.y;
    bv.z = b0.z * be.z; bv.w = b0.w * be.w;

    // -------- streaming setup ---------------------------------------------
    const int hin0    = strip * RSTRIP;    // first input row of this strip
    const int in_rows = RSTRIP + 2;        // 29 input rows per strip

    const float* Xn = X + (size_t)n * HDIM * WDIM * CDIM + (size_t)cg * CGRP;

    // One input row = 3584 floats = 896 float4. Every thread issues exactly
    // 4 async DMA ops (3x b128 covering floats [0,3072) + 1x b64 covering
    // [3072,3584)) so ASYNCcnt is uniform (4/row) across all 8 waves.
    auto issue_row = [&](int r) {
        const float*   rowg  = Xn + (size_t)(hin0 + r) * WDIM * CDIM;
        const uint32_t slotb = lds_base + (uint32_t)((r & 3) * ROWF * 4);
#pragma unroll
        for (int k = 0; k < 3; ++k) {
            const int v  = tid + k * 256;      // float4 id in row, [0,768)
            const int w  = v >> 4;             // column 0..47
            const int cv = v & 15;             // float4 within 64-ch column
            async_b128(slotb + (uint32_t)((w * CGRP + cv * 4) * 4),
                       rowg + (size_t)w * CDIM + cv * 4);
        }
        {
            const int f = 3072 + tid * 2;      // float id, [3072,3584)
            const int w = f >> 6;              // column 48..55
            const int c = f & 63;              // even -> 8B aligned
            async_b64(slotb + (uint32_t)((w * CGRP + c) * 4),
                      rowg + (size_t)w * CDIM + c);
        }
    };

    issue_row(0); issue_row(1); issue_row(2);   // prologue: 12 ops in flight

    float* outn = out + (size_t)n * HOUT * WOUT * CDIM + (size_t)cg * CGRP;

    for (int h = 0; h < RSTRIP; ++h) {
        __syncthreads();                       // slot (h+3)&3 no longer read
        if (h + 3 < in_rows) {
            issue_row(h + 3);                  // prefetch depth 1
            wait_async_le4();                  // rows <= h+2 complete
        } else {
            wait_async_0();                    // drain tail
        }
        __syncthreads();                       // LDS visible to all waves

        const int ho = hin0 + h;
        const float4* s0 = reinterpret_cast<const float4*>(&smem[((h + 0) & 3) * ROWF]);
        const float4* s1 = reinterpret_cast<const float4*>(&smem[((h + 1) & 3) * ROWF]);
        const float4* s2 = reinterpret_cast<const float4*>(&smem[((h + 2) & 3) * ROWF]);

#pragma unroll
        for (int wb = 0; wb < 64; wb += 16) {
            const int wo = wb + (tid >> 4);
            if (wo < WOUT) {
                float4 acc = bv;
#pragma unroll
                for (int kw = 0; kw < 3; ++kw) {
                    const float4 x0 = s0[(wo + kw) * 16 + c4];
                    const float4 x1 = s1[(wo + kw) * 16 + c4];
                    const float4 x2 = s2[(wo + kw) * 16 + c4];
                    acc.x = fmaf(x0.x, wv[0 + kw].x, acc.x);
                    acc.y = fmaf(x0.y, wv[0 + kw].y, acc.y);
                    acc.z = fmaf(x0.z, wv[0 + kw].z, acc.z);
                    acc.w = fmaf(x0.w, wv[0 + kw].w, acc.w);
                    acc.x = fmaf(x1.x, wv[3 + kw].x, acc.x);
                    acc.y = fmaf(x1.y, wv[3 + kw].y, acc.y);
                    acc.z = fmaf(x1.z, wv[3 + kw].z, acc.z);
                    acc.w = fmaf(x1.w, wv[3 + kw].w, acc.w);
                    acc.x = fmaf(x2.x, wv[6 + kw].x, acc.x);
                    acc.y = fmaf(x2.y, wv[6 + kw].y, acc.y);
                    acc.z = fmaf(x2.z, wv[6 + kw].z, acc.z);
                    acc.w = fmaf(x2.w, wv[6 + kw].w, acc.w);
                }
                acc.x = quant8(acc.x); acc.y = quant8(acc.y);
                acc.z = quant8(acc.z); acc.w = quant8(acc.w);
                reinterpret_cast<float4*>(outn + ((size_t)ho * WOUT + wo) * CDIM)[c4] = acc;
            }
        }
    }
}

extern "C" void kernel_launch(void* const* d_in, const int* in_sizes, int n_in,
                              void* d_out, int out_size, void* d_ws, size_t ws_size,
                              hipStream_t stream) {
    const float* X    = (const float*)d_in[0];
    const float* W    = (const float*)d_in[1];
    const float* bias = (const float*)d_in[2];
    const float* Werr = (const float*)d_in[3];
    const float* Berr = (const float*)d_in[4];
    float* out = (float*)d_out;

    dim3 grid(STRIPS, NCGRP, 64);   // 2 x 4 x 64 = 512 blocks
    dwconv_aconnect_kernel<<<grid, 256, 0, stream>>>(X, W, bias, Werr, Berr, out);
}